// MambaAggregator1D_48060684042616
// MI455X (gfx1250) — compile-verified
//
#include <hip/hip_runtime.h>
#include <math.h>

// ---------------- problem constants ----------------
constexpr int kB  = 4;
constexpr int kL  = 1024;
constexpr int kDM = 1024;
constexpr int kDI = 2048;
constexpr int kN  = 16;
constexpr int kR  = 64;

constexpr int M1 = kB * kL;   // 4096 rows
constexpr int N1 = 2 * kDI;   // 4096 cols (xi | z)
constexpr int K1 = kDM;       // 1024
constexpr int NCHUNK = K1 / 32;

typedef __attribute__((ext_vector_type(2))) float v2f;
typedef __attribute__((ext_vector_type(8))) float v8f;
typedef __attribute__((ext_vector_type(4))) int   v4i;

__device__ __forceinline__ float silu_f(float x) {
    return x / (1.0f + __expf(-x));
}

// -------- CDNA5 async global->LDS copy (ASYNCcnt path) with fallback --------
#if defined(__HIP_DEVICE_COMPILE__) && __has_builtin(__builtin_amdgcn_global_load_async_to_lds_b128)
#define ASYNC_LDS 1
#else
#define ASYNC_LDS 0
#endif

__device__ __forceinline__ void copy16_g2l(const float* __restrict__ g, float* l) {
#if ASYNC_LDS
    __builtin_amdgcn_global_load_async_to_lds_b128(
        (__attribute__((address_space(1))) v4i*)g,
        (__attribute__((address_space(3))) v4i*)l,
        /*offset=*/0, /*cpol=*/0);
#else
    *(float4*)l = *(const float4*)g;
#endif
}

__device__ __forceinline__ void wait_async0() {
#if ASYNC_LDS
#if __has_builtin(__builtin_amdgcn_s_wait_asynccnt)
    __builtin_amdgcn_s_wait_asynccnt(0);
#else
    asm volatile("s_wait_asynccnt 0x0" ::: "memory");
#endif
#endif
}

// =====================================================================
// Kernel 1: xz = X @ W_in, fused split + silu on the z half.
// WMMA f32 16x16x4, block tile 64(M) x 128(N), K-chunk 32, double-buffered
// LDS staged via GLOBAL_LOAD_ASYNC_TO_LDS_B128 (overlaps with WMMA).
// 256 threads = 8 waves; wave w -> M sub-tile (w&3), N half ((w>>2)*64).
// =====================================================================
__device__ __forceinline__ void stage_chunk(const float* __restrict__ X,
                                            const float* __restrict__ Win,
                                            float (*As)[36], float (*Bs)[136],
                                            int mBlock, int nBlock, int k0, int t)
{
    // A tile: 64 rows x 32 K (2 x b128 per thread)
    const int rowA = t >> 3;
    const int tq   = t & 7;
#pragma unroll
    for (int p = 0; p < 2; ++p) {
        const int r = rowA + p * 32;
        copy16_g2l(X + (size_t)(mBlock + r) * K1 + k0 + tq * 4, &As[r][tq * 4]);
    }
    // B tile: 32 rows x 128 N (4 x b128 per thread)
    const int cq = t & 31;
    const int r8 = t >> 5;
#pragma unroll
    for (int p = 0; p < 4; ++p) {
        const int r = p * 8 + r8;
        copy16_g2l(Win + (size_t)(k0 + r) * N1 + nBlock + cq * 4, &Bs[r][cq * 4]);
    }
}

__launch_bounds__(256)
__global__ void gemm_in_kernel(const float* __restrict__ X,
                               const float* __restrict__ Win,
                               float* __restrict__ xi,
                               float* __restrict__ zact)
{
    __shared__ float As[2][64][36];    // pad 36: 16B rows, conflict-free frags
    __shared__ float Bs[2][32][136];   // pad 136: 16B rows, disjoint half-wave banks

    const int t    = threadIdx.x;
    const int wave = t >> 5;
    const int lane = t & 31;
    const int half = lane >> 4;     // 0: lanes 0-15, 1: lanes 16-31
    const int l16  = lane & 15;

    const int mBlock = blockIdx.y * 64;
    const int nBlock = blockIdx.x * 128;
    const int mr = wave & 3;          // M sub-tile 0..3
    const int nc = (wave >> 2) * 64;  // N offset 0 or 64

    v8f acc[4];
#pragma unroll
    for (int j = 0; j < 4; ++j)
#pragma unroll
        for (int e = 0; e < 8; ++e) acc[j][e] = 0.0f;

    // prologue: stage chunk 0 into buffer 0
    stage_chunk(X, Win, As[0], Bs[0], mBlock, nBlock, 0, t);

    for (int i = 0; i < NCHUNK; ++i) {
        const int p = i & 1;
        wait_async0();          // chunk i resident in LDS (this wave's copies)
        __syncthreads();        // visible to all waves; prior reads of buf 1-p done
        if (i + 1 < NCHUNK)     // overlap: stream chunk i+1 while computing chunk i
            stage_chunk(X, Win, As[1 - p], Bs[1 - p], mBlock, nBlock, (i + 1) * 32, t);

        float (*Ac)[36]  = As[p];
        float (*Bc)[136] = Bs[p];
#pragma unroll
        for (int kk = 0; kk < 32; kk += 4) {
            // A 16x4 frag: lane half selects K {0,1} vs {2,3}
            v2f a;
            a.x = Ac[mr * 16 + l16][kk + 2 * half];
            a.y = Ac[mr * 16 + l16][kk + 2 * half + 1];
#pragma unroll
            for (int j = 0; j < 4; ++j) {
                v2f b;
                b.x = Bc[kk + 2 * half    ][nc + j * 16 + l16];
                b.y = Bc[kk + 2 * half + 1][nc + j * 16 + l16];
                acc[j] = __builtin_amdgcn_wmma_f32_16x16x4_f32(
                    /*neg_a=*/false, a, /*neg_b=*/false, b,
                    /*c_mod=*/(short)0, acc[j],
                    /*reuse_a=*/false, /*reuse_b=*/false);
            }
        }
    }

    // ---- epilogue: split columns, silu on z half (uniform per 16-wide tile) ----
#pragma unroll
    for (int j = 0; j < 4; ++j) {
        const int col = nBlock + nc + j * 16 + l16;
#pragma unroll
        for (int vv = 0; vv < 8; ++vv) {
            const int row = mBlock + mr * 16 + vv + 8 * half;
            const float val = acc[j][vv];
            if (col < kDI) {
                xi[(size_t)row * kDI + col] = val;
            } else {
                zact[(size_t)row * kDI + (col - kDI)] = silu_f(val);
            }
        }
    }
}

// =====================================================================
// Kernel 2: depthwise conv width-3 (same padding) along L + bias + silu
// =====================================================================
__launch_bounds__(256)
__global__ void conv_silu_kernel(const float* __restrict__ xi,
                                 const float* __restrict__ cw,
                                 const float* __restrict__ cb,
                                 float* __restrict__ u)
{
    const size_t total = (size_t)kB * kL * kDI;
    size_t idx = (size_t)blockIdx.x * blockDim.x + threadIdx.x;
    if (idx >= total) return;
    const int d = (int)(idx % kDI);
    const size_t bl = idx / kDI;
    const int l = (int)(bl % kL);

    const float w0 = cw[d * 3 + 0];
    const float w1 = cw[d * 3 + 1];
    const float w2 = cw[d * 3 + 2];

    const float* base = xi + bl * kDI + d;
    float acc = cb[d] + base[0] * w1;
    if (l > 0)      acc += base[-(ptrdiff_t)kDI] * w0;
    if (l < kL - 1) acc += base[kDI] * w2;
    u[idx] = silu_f(acc);
}

// =====================================================================
// Kernel 3: per row (b,l): x_dbl = u_row @ W_x  (96 cols),
//           delta = softplus(x_dbl[:64] @ W_dt + dt_bias), Bs/Cs slices
// =====================================================================
__launch_bounds__(256)
__global__ void xdbl_kernel(const float* __restrict__ u,
                            const float* __restrict__ Wx,
                            const float* __restrict__ Wdt,
                            const float* __restrict__ dt_bias,
                            float* __restrict__ delta,
                            float* __restrict__ Bs_o,
                            float* __restrict__ Cs_o)
{
    __shared__ float su[kDI];    // 8 KB
    __shared__ float sxd[96];

    const int row = blockIdx.x;   // 0..4095
    const int t = threadIdx.x;

    const float* urow = u + (size_t)row * kDI;
    for (int i = t; i < kDI; i += 256) su[i] = urow[i];
    __syncthreads();

    if (t < 96) {
        float acc = 0.0f;
#pragma unroll 4
        for (int k = 0; k < kDI; ++k) acc += su[k] * Wx[(size_t)k * 96 + t];
        sxd[t] = acc;
    }
    __syncthreads();

    if (t < 16)       Bs_o[(size_t)row * kN + t]        = sxd[kR + t];
    else if (t < 32)  Cs_o[(size_t)row * kN + (t - 16)] = sxd[kR + kN + (t - 16)];

    for (int d = t; d < kDI; d += 256) {
        float acc = dt_bias[d];
#pragma unroll 8
        for (int r = 0; r < kR; ++r) acc += sxd[r] * Wdt[(size_t)r * kDI + d];
        delta[(size_t)row * kDI + d] = (acc > 20.0f) ? acc : log1pf(__expf(acc));
    }
}

// =====================================================================
// Kernel 4: zero the v accumulator
// =====================================================================
__global__ void zero_kernel(float* __restrict__ p, int n)
{
    int i = blockIdx.x * blockDim.x + threadIdx.x;
    if (i < n) p[i] = 0.0f;
}

// =====================================================================
// Kernel 5: sequential SSM scan. One thread per (b,d); h[16] in regs;
// Bs/Cs (32 floats/step) broadcast via double-buffered LDS -> 1 barrier/step.
// Folds + Ds*u into y.
// =====================================================================
__launch_bounds__(256)
__global__ void scan_kernel(const float* __restrict__ delta,
                            const float* __restrict__ u,
                            const float* __restrict__ Bs_i,
                            const float* __restrict__ Cs_i,
                            const float* __restrict__ A_logs,
                            const float* __restrict__ Ds,
                            float* __restrict__ y)
{
    __shared__ float sbc[2][32];
    const int tid = threadIdx.x;
    const int d = blockIdx.x * 256 + tid;
    const int b = blockIdx.y;

    float A[kN];
#pragma unroll
    for (int n = 0; n < kN; ++n) A[n] = -__expf(A_logs[(size_t)d * kN + n]);
    float h[kN];
#pragma unroll
    for (int n = 0; n < kN; ++n) h[n] = 0.0f;
    const float Dd = Ds[d];

    // preload step 0 broadcast data
    if (tid < 32) {
        const size_t rb0 = (size_t)b * kL;
        sbc[0][tid] = (tid < 16) ? Bs_i[rb0 * kN + tid]
                                 : Cs_i[rb0 * kN + (tid - 16)];
    }
    __syncthreads();

    for (int l = 0; l < kL; ++l) {
        const int p = l & 1;
        const size_t rb = (size_t)b * kL + l;
        // prefetch step l+1 into the other buffer (no conflict with readers of p)
        if (tid < 32 && l + 1 < kL) {
            const size_t rb1 = rb + 1;
            sbc[1 - p][tid] = (tid < 16) ? Bs_i[rb1 * kN + tid]
                                         : Cs_i[rb1 * kN + (tid - 16)];
        }

        const float dt = delta[rb * kDI + d];
        const float ut = u[rb * kDI + d];
        const float du = dt * ut;
        float acc = 0.0f;
#pragma unroll
        for (int n = 0; n < kN; ++n) {
            h[n] = __expf(dt * A[n]) * h[n] + du * sbc[p][n];
            acc += h[n] * sbc[p][16 + n];
        }
        y[rb * kDI + d] = acc + Dd * ut;
        __syncthreads();   // readers of buf p done; writers of buf p (next iter) may proceed
    }
}

// =====================================================================
// Kernel 6: per-row LayerNorm over DI, * z, accumulate sum over L into v
// =====================================================================
__launch_bounds__(256)
__global__ void norm_kernel(const float* __restrict__ y,
                            const float* __restrict__ z,
                            const float* __restrict__ ln_g,
                            const float* __restrict__ ln_b,
                            float* __restrict__ v)
{
    __shared__ float red[256];
    const int row = blockIdx.x;        // b*L + l
    const int b = row / kL;
    const int t = threadIdx.x;

    const float* yr = y + (size_t)row * kDI;
    float vals[8];
    float s = 0.0f;
#pragma unroll
    for (int i = 0; i < 8; ++i) { vals[i] = yr[t + i * 256]; s += vals[i]; }

    red[t] = s; __syncthreads();
    for (int st = 128; st > 0; st >>= 1) { if (t < st) red[t] += red[t + st]; __syncthreads(); }
    const float mu = red[0] * (1.0f / kDI);
    __syncthreads();

    float ss = 0.0f;
#pragma unroll
    for (int i = 0; i < 8; ++i) { const float dv = vals[i] - mu; ss += dv * dv; }
    red[t] = ss; __syncthreads();
    for (int st = 128; st > 0; st >>= 1) { if (t < st) red[t] += red[t + st]; __syncthreads(); }
    const float rstd = rsqrtf(red[0] * (1.0f / kDI) + 1e-5f);

#pragma unroll
    for (int i = 0; i < 8; ++i) {
        const int dc = t + i * 256;
        float o = (vals[i] - mu) * rstd * ln_g[dc] + ln_b[dc];
        o *= z[(size_t)row * kDI + dc];
        atomicAdd(&v[(size_t)b * kDI + dc], o);
    }
}

// =====================================================================
// Kernel 7: out = (v / L) @ W_out + b_out   (4 x 2048) @ (2048 x 1024)
// =====================================================================
__launch_bounds__(256)
__global__ void out_kernel(const float* __restrict__ v,
                           const float* __restrict__ Wout,
                           const float* __restrict__ bout,
                           float* __restrict__ out)
{
    const int idx = blockIdx.x * 256 + threadIdx.x;
    if (idx >= kB * kDM) return;
    const int b = idx / kDM;
    const int m = idx % kDM;
    const float invL = 1.0f / kL;
    const float* vr = v + (size_t)b * kDI;
    float acc = bout[m];
    for (int k = 0; k < kDI; ++k)
        acc += (vr[k] * invL) * Wout[(size_t)k * kDM + m];
    out[idx] = acc;
}

// =====================================================================
extern "C" void kernel_launch(void* const* d_in, const int* in_sizes, int n_in,
                              void* d_out, int out_size, void* d_ws, size_t ws_size,
                              hipStream_t stream)
{
    const float* x       = (const float*)d_in[0];
    const float* W_in    = (const float*)d_in[1];
    const float* conv_w  = (const float*)d_in[2];
    const float* conv_b  = (const float*)d_in[3];
    const float* W_x     = (const float*)d_in[4];
    const float* W_dt    = (const float*)d_in[5];
    const float* dt_bias = (const float*)d_in[6];
    const float* A_logs  = (const float*)d_in[7];
    const float* Ds      = (const float*)d_in[8];
    const float* ln_g    = (const float*)d_in[9];
    const float* ln_b    = (const float*)d_in[10];
    const float* W_out   = (const float*)d_in[11];
    const float* b_out   = (const float*)d_in[12];
    float* out = (float*)d_out;

    // workspace layout (floats); xi is dead after conv -> delta aliases it
    const size_t big = (size_t)M1 * kDI;           // 8M floats
    float* ws   = (float*)d_ws;
    float* xi    = ws;                              // [0, big)
    float* delta = ws;                              // alias of xi
    float* zbuf  = ws + big;                        // [big, 2big)
    float* ubuf  = ws + 2 * big;                    // [2big, 3big)
    float* ybuf  = ws + 3 * big;                    // [3big, 4big)
    float* Bs    = ws + 4 * big;                    // 4096*16
    float* Cs    = Bs + (size_t)M1 * kN;
    float* vacc  = Cs + (size_t)M1 * kN;            // 4*2048

    // 1) big GEMM (WMMA f32, async double-buffered staging) + split/silu
    dim3 g1(N1 / 128, M1 / 64);
    gemm_in_kernel<<<g1, 256, 0, stream>>>(x, W_in, xi, zbuf);

    // 2) depthwise conv + silu -> u
    const size_t total = (size_t)kB * kL * kDI;
    conv_silu_kernel<<<(unsigned)((total + 255) / 256), 256, 0, stream>>>(xi, conv_w, conv_b, ubuf);

    // 3) x_dbl projections -> delta (overwrites xi), Bs, Cs
    xdbl_kernel<<<M1, 256, 0, stream>>>(ubuf, W_x, W_dt, dt_bias, delta, Bs, Cs);

    // 4) zero v accumulator
    zero_kernel<<<(kB * kDI + 255) / 256, 256, 0, stream>>>(vacc, kB * kDI);

    // 5) sequential scan -> y (+ Ds*u folded)
    dim3 g5(kDI / 256, kB);
    scan_kernel<<<g5, 256, 0, stream>>>(delta, ubuf, Bs, Cs, A_logs, Ds, ybuf);

    // 6) layernorm * z, accumulate L-sum into v
    norm_kernel<<<M1, 256, 0, stream>>>(ybuf, zbuf, ln_g, ln_b, vacc);

    // 7) final projection
    out_kernel<<<(kB * kDM + 255) / 256, 256, 0, stream>>>(vacc, W_out, b_out, out);
}